// MultiHeadSelfAttention2d_33011118637367
// MI455X (gfx1250) — compile-verified
//
#include <hip/hip_runtime.h>
#include <math.h>

// MI455X / gfx1250, wave32. All matrix math via V_WMMA_F32_16X16X4_F32 (full f32,
// matches the f32 reference). Flash-attention fusion avoids the 512MB attn matrix.
// K/V tiles are staged per-block into LDS with GLOBAL_LOAD_ASYNC_TO_LDS_B32
// (ASYNCcnt-tracked, double-buffered) and shared by the 4 waves of a block.

typedef float v2f __attribute__((ext_vector_type(2)));
typedef float v8f __attribute__((ext_vector_type(8)));

#define N_PIX 4096
#define C_IN 64
#define HEADS 8
#define HIDDEN 16
#define HD 128
#define H_IMG 32
#define W_IMG 128
#define OUT_DIM 64

__device__ __forceinline__ v8f wmma_f32(v2f a, v2f b, v8f c) {
  // (neg_a, A, neg_b, B, c_mod, C, reuse_a, reuse_b)
  return __builtin_amdgcn_wmma_f32_16x16x4_f32(false, a, false, b, (short)0, c,
                                               false, false);
}

// ---------------------------------------------------------------------------
// Stage 1: 1x1 conv (per-pixel linear) as GEMM: W[128x64] * X[64x4096] + bias.
// vmode==0 -> store [o][n] (channel-major, feeds K staging contiguously)
// vmode==1 -> store V transposed [head][n][16] (contiguous V-tile staging)
// ---------------------------------------------------------------------------
__global__ void qkv_proj_kernel(const float* __restrict__ x,
                                const float* __restrict__ w,
                                const float* __restrict__ bias,
                                float* __restrict__ out, int vmode) {
  const int lane = threadIdx.x & 31;
  const int lo = lane & 15;
  const int hi = lane >> 4;
  const int wave = blockIdx.x * (blockDim.x >> 5) + (threadIdx.x >> 5);
  const int otile = wave >> 8;   // 0..7  (128 output channels / 16)
  const int ntile = wave & 255;  // 0..255 (4096 pixels / 16)
  const int obase = otile * 16;
  const int nbase = ntile * 16;

  v8f acc = {};
#pragma unroll
  for (int kc = 0; kc < C_IN / 4; ++kc) {
    const int k0 = kc * 4 + hi * 2;
    v2f a, b;
    a[0] = w[(obase + lo) * C_IN + k0];
    a[1] = w[(obase + lo) * C_IN + k0 + 1];
    b[0] = x[k0 * N_PIX + nbase + lo];  // coalesced over lo
    b[1] = x[(k0 + 1) * N_PIX + nbase + lo];
    acc = wmma_f32(a, b, acc);
  }
#pragma unroll
  for (int r = 0; r < 8; ++r) {
    const int o = obase + r + 8 * hi;  // C/D row M
    const int n = nbase + lo;          // C/D col
    const float val = acc[r] + bias[o];
    if (vmode == 0) {
      out[o * N_PIX + n] = val;
    } else {
      out[(o >> 4) * (N_PIX * HIDDEN) + n * HIDDEN + (o & 15)] = val;
    }
  }
}

// ---------------------------------------------------------------------------
// Stage 2: fused flash attention. Block = 4 waves, all on the SAME head so the
// 16x16 K and V tiles are staged once per block via async LDS copies and
// shared. One wave = one 16-row Q tile; 256 key tiles, online softmax.
// ---------------------------------------------------------------------------
__global__ void attn_kernel(const float* __restrict__ q,   // [128][N]
                            const float* __restrict__ k,   // [128][N]
                            const float* __restrict__ vt,  // [8][N][16]
                            float* __restrict__ ao) {      // [4096][128]
  __shared__ float kTile[2][256];  // [d][m'] per buffer
  __shared__ float vTile[2][256];  // [m'][d] per buffer
  __shared__ float pLds[4][256];   // per-wave P tile round-trip

  const int t = threadIdx.x;
  const int lane = t & 31;
  const int lo = lane & 15;
  const int hi = lane >> 4;
  const int warp = t >> 5;
  const int head = blockIdx.x >> 6;                 // 64 blocks per head
  const int ntile = (blockIdx.x & 63) * 4 + warp;   // 0..255
  const int nbase = ntile * 16;
  const float scale = 0.25f;  // 1/sqrt(16)

  const float* khead = k + (size_t)head * HIDDEN * N_PIX;   // [d][m]
  const float* vhead = vt + (size_t)head * N_PIX * HIDDEN;  // [m][d]

  // LDS byte offsets (flat LDS aperture: low 32 bits == wave LDS offset)
  const unsigned kT = (unsigned)(uintptr_t)&kTile[0][0];
  const unsigned vT = (unsigned)(uintptr_t)&vTile[0][0];

  // Each thread async-copies 2 K floats + 2 V floats per tile (4 async/wave).
  const int idx0 = t;        // 0..127
  const int idx1 = t + 128;  // 128..255

  // Q tile as 4 A-fragments (K-chunks of 4 over HIDDEN=16), kept in VGPRs.
  v2f qa[4];
#pragma unroll
  for (int kc = 0; kc < 4; ++kc) {
    const int d = kc * 4 + hi * 2;
    qa[kc][0] = q[(head * HIDDEN + d) * N_PIX + nbase + lo];
    qa[kc][1] = q[(head * HIDDEN + d + 1) * N_PIX + nbase + lo];
  }

  float mrow[8], lrow[8];
#pragma unroll
  for (int r = 0; r < 8; ++r) { mrow[r] = -3.0e38f; lrow[r] = 0.0f; }
  v8f acc = {};  // O tile: row = query n, col = d

  // --- async staging of one (K,V) tile pair into buffer `buf` ---
  auto stage_tiles = [&](int mbase, int buf) {
    const unsigned kdst = kT + (unsigned)(buf * 1024);
    const unsigned vdst = vT + (unsigned)(buf * 1024);
    const unsigned ko0 = (unsigned)(((idx0 >> 4) * N_PIX + mbase + (idx0 & 15)) * 4);
    const unsigned ko1 = (unsigned)(((idx1 >> 4) * N_PIX + mbase + (idx1 & 15)) * 4);
    const unsigned vo0 = (unsigned)((mbase * HIDDEN + idx0) * 4);
    const unsigned vo1 = (unsigned)((mbase * HIDDEN + idx1) * 4);
    asm volatile("global_load_async_to_lds_b32 %0, %1, %2" ::"v"(kdst + idx0 * 4),
                 "v"(ko0), "s"((unsigned long long)(uintptr_t)khead)
                 : "memory");
    asm volatile("global_load_async_to_lds_b32 %0, %1, %2" ::"v"(kdst + idx1 * 4),
                 "v"(ko1), "s"((unsigned long long)(uintptr_t)khead)
                 : "memory");
    asm volatile("global_load_async_to_lds_b32 %0, %1, %2" ::"v"(vdst + idx0 * 4),
                 "v"(vo0), "s"((unsigned long long)(uintptr_t)vhead)
                 : "memory");
    asm volatile("global_load_async_to_lds_b32 %0, %1, %2" ::"v"(vdst + idx1 * 4),
                 "v"(vo1), "s"((unsigned long long)(uintptr_t)vhead)
                 : "memory");
  };

  stage_tiles(0, 0);  // prefetch first tile

  for (int mtile = 0; mtile < N_PIX / 16; ++mtile) {
    const int buf = mtile & 1;
    if (mtile + 1 < N_PIX / 16) {
      stage_tiles((mtile + 1) * 16, buf ^ 1);  // prefetch next (4 newer ops)
      asm volatile("s_wait_asynccnt 0x4" ::: "memory");  // drain current tile
    } else {
      asm volatile("s_wait_asynccnt 0x0" ::: "memory");
    }
    __syncthreads();  // tile visible to all 4 waves

    // S = Q K^T for this 16-key tile (B-frags from LDS kTile)
    v8f s = {};
#pragma unroll
    for (int kc = 0; kc < 4; ++kc) {
      const int d0 = kc * 4 + hi * 2;
      v2f kb;
      kb[0] = kTile[buf][d0 * 16 + lo];
      kb[1] = kTile[buf][(d0 + 1) * 16 + lo];
      s = wmma_f32(qa[kc], kb, s);
    }
    // Online softmax: row M = r + 8*hi lives across the 16-lane column group.
#pragma unroll
    for (int r = 0; r < 8; ++r) {
      float sv = s[r] * scale;
      float rmax = sv;
      rmax = fmaxf(rmax, __shfl_xor(rmax, 1));
      rmax = fmaxf(rmax, __shfl_xor(rmax, 2));
      rmax = fmaxf(rmax, __shfl_xor(rmax, 4));
      rmax = fmaxf(rmax, __shfl_xor(rmax, 8));
      const float newm = fmaxf(mrow[r], rmax);
      const float corr = __expf(mrow[r] - newm);
      const float p = __expf(sv - newm);
      float rsum = p;
      rsum += __shfl_xor(rsum, 1);
      rsum += __shfl_xor(rsum, 2);
      rsum += __shfl_xor(rsum, 4);
      rsum += __shfl_xor(rsum, 8);
      lrow[r] = lrow[r] * corr + rsum;
      mrow[r] = newm;
      acc[r] *= corr;  // rescale running output (same row mapping as S)
      pLds[warp][(r + 8 * hi) * 16 + lo] = p;  // C/D layout -> LDS
    }
    asm volatile("s_wait_dscnt 0x0" ::: "memory");  // P stores -> A-frag loads
#pragma unroll
    for (int kc = 0; kc < 4; ++kc) {
      const int m0 = kc * 4 + hi * 2;
      v2f pa, vb;
      pa[0] = pLds[warp][lo * 16 + m0];  // A-frag of P: row=lo, K=m
      pa[1] = pLds[warp][lo * 16 + m0 + 1];
      vb[0] = vTile[buf][m0 * 16 + lo];
      vb[1] = vTile[buf][(m0 + 1) * 16 + lo];
      acc = wmma_f32(pa, vb, acc);
    }
    __syncthreads();  // all readers done before this buffer is re-staged
  }

  // Normalize and scatter to [R = ch*32 + h][w] layout for the final linear.
#pragma unroll
  for (int r = 0; r < 8; ++r) {
    const int n = nbase + r + 8 * hi;
    const int ch = head * HIDDEN + lo;
    const int hh = n >> 7;   // n / 128
    const int wi = n & 127;  // n % 128
    ao[(ch * H_IMG + hh) * W_IMG + wi] = acc[r] / lrow[r];
  }
}

// ---------------------------------------------------------------------------
// Stage 3: final linear over the W axis: Y[4096x64] = AO[4096x128] * Wl^T + b.
// ---------------------------------------------------------------------------
__global__ void final_linear_kernel(const float* __restrict__ ao,
                                    const float* __restrict__ wl,  // [64][128]
                                    const float* __restrict__ bl,  // [64]
                                    float* __restrict__ y) {       // [4096][64]
  const int lane = threadIdx.x & 31;
  const int lo = lane & 15;
  const int hi = lane >> 4;
  const int wave = blockIdx.x * (blockDim.x >> 5) + (threadIdx.x >> 5);
  const int rtile = wave >> 2;  // 0..255
  const int otile = wave & 3;   // 0..3
  v8f acc = {};
#pragma unroll
  for (int kc = 0; kc < HD / 4; ++kc) {
    const int k0 = kc * 4 + hi * 2;
    v2f a, b;
    a[0] = ao[(rtile * 16 + lo) * HD + k0];
    a[1] = ao[(rtile * 16 + lo) * HD + k0 + 1];
    b[0] = wl[(otile * 16 + lo) * HD + k0];  // B[k][o] = wl[o][k]
    b[1] = wl[(otile * 16 + lo) * HD + k0 + 1];
    acc = wmma_f32(a, b, acc);
  }
#pragma unroll
  for (int r = 0; r < 8; ++r) {
    const int R = rtile * 16 + r + 8 * hi;
    const int o = otile * 16 + lo;
    y[R * OUT_DIM + o] = acc[r] + bl[o];
  }
}

// ---------------------------------------------------------------------------
extern "C" void kernel_launch(void* const* d_in, const int* in_sizes, int n_in,
                              void* d_out, int out_size, void* d_ws,
                              size_t ws_size, hipStream_t stream) {
  const float* x = (const float*)d_in[0];
  const float* wq = (const float*)d_in[1];
  const float* bq = (const float*)d_in[2];
  const float* wk = (const float*)d_in[3];
  const float* bk = (const float*)d_in[4];
  const float* wv = (const float*)d_in[5];
  const float* bv = (const float*)d_in[6];
  const float* wl = (const float*)d_in[7];
  const float* bl = (const float*)d_in[8];
  float* y = (float*)d_out;

  float* ws = (float*)d_ws;
  const size_t seg = (size_t)HD * N_PIX;  // 524288 floats per buffer
  float* qbuf = ws;
  float* kbuf = ws + seg;
  float* vbuf = ws + 2 * seg;
  float* aobuf = ws + 3 * seg;

  // 2048 tile-waves per projection -> 512 blocks x 4 waves
  qkv_proj_kernel<<<512, 128, 0, stream>>>(x, wq, bq, qbuf, 0);
  qkv_proj_kernel<<<512, 128, 0, stream>>>(x, wk, bk, kbuf, 0);
  qkv_proj_kernel<<<512, 128, 0, stream>>>(x, wv, bv, vbuf, 1);
  // 8 heads x 64 blocks, 4 waves/block all on the same head
  attn_kernel<<<512, 128, 0, stream>>>(qbuf, kbuf, vbuf, aobuf);
  // 256 row-tiles x 4 col-tiles = 1024 waves
  final_linear_kernel<<<256, 128, 0, stream>>>(aobuf, wl, bl, y);
}